// AdditiveAttn_33792802685123
// MI455X (gfx1250) — compile-verified
//
#include <hip/hip_runtime.h>

// ---------------------------------------------------------------------------
// Additive graph attention (GATv2-style) fully fused for gfx1250 (MI455X).
//
//   K1: Nq/Nk/Nv = x @ W{q,k,v} + b                      -- WMMA f32 16x16x4
//   K2 (single fused edge pass, 64 edges / workgroup):
//        Eq   = edge_attr @ Weq + beq                    -- WMMA, B-frag reused
//        conn = relu(Nk[src] + Nq[dst] + Eq)  -> e_out
//        s    = clamp(conn . Aw, -5, 5); p = exp(s)
//        (max-subtraction is omitted: s is clamped to [-5,5], so exp(s) is
//         bounded in [6.7e-3, 148] and exp(s)/Σexp(s) == exp(s-m)/Σexp(s-m))
//        atomic: ssum[dst] += p; agg1[dst] += p*Nv[src]; agg2[dst] += p*conn
//        (all scatter targets are L2-resident: N*HD*4 = 25.6 MB each)
//   K3: n_out = (agg1 + agg2 @ Ew_head) / (ssum + 1e-16) -- WMMA f32 16x16x4
// ---------------------------------------------------------------------------

typedef float v2f __attribute__((ext_vector_type(2)));
typedef float v8f __attribute__((ext_vector_type(8)));

#define N_HEADS 8
#define HDIM    16
#define HD      128   // HEADS*HDIM
#define INDIM   128
#define LDSPITCH (INDIM + 4)  // 132: 132%64=4 -> conflict-free column access
#define CLAMP_V 5.0f
#define EPB     64            // edges per block (4 M-tiles of 16)

__device__ __forceinline__ v8f wmma_f32_16x16x4(v2f a, v2f b, v8f c) {
  // 8 args: (neg_a, A, neg_b, B, c_mod, C, reuse_a, reuse_b)
  return __builtin_amdgcn_wmma_f32_16x16x4_f32(false, a, false, b, (short)0, c,
                                               false, false);
}

// --------------------------- K1: node Q/K/V GEMM ---------------------------
// 16 nodes per block; one wave per 16-column slab; x tile staged in LDS.
// A layout: lanes 0-15 -> M=lane, K={k0,k0+1}; lanes 16-31 -> M=lane-16,
//           K={k0+2,k0+3}.  B mirrors with N=lane%16.
__global__ __launch_bounds__(256) void qkv_gemm_kernel(
    const float* __restrict__ x, const float* __restrict__ Wq,
    const float* __restrict__ bq, const float* __restrict__ Wk,
    const float* __restrict__ bk, const float* __restrict__ Wv,
    const float* __restrict__ bv, float* __restrict__ Nq,
    float* __restrict__ Nk, float* __restrict__ Nv) {
  __shared__ float xs[16][LDSPITCH];
  const int tid = threadIdx.x;
  const int m0 = blockIdx.x * 16;
  {
    const int row = tid >> 4;
    const int c0 = (tid & 15) * 8;
    const float* src = x + (size_t)(m0 + row) * INDIM + c0;
#pragma unroll
    for (int j = 0; j < 8; ++j) xs[row][c0 + j] = src[j];
  }
  __syncthreads();

  const int lane = tid & 31;
  const int half = lane >> 4;
  const int ln = lane & 15;
  const int col0 = (tid >> 5) * 16;

  const float* Ws[3] = {Wq, Wk, Wv};
  const float* bs[3] = {bq, bk, bv};
  float* outs[3] = {Nq, Nk, Nv};
#pragma unroll
  for (int m = 0; m < 3; ++m) {
    const float* W = Ws[m];
    const float bias = bs[m][col0 + ln];
    v8f c;
#pragma unroll
    for (int r = 0; r < 8; ++r) c[r] = bias;
#pragma unroll
    for (int k0 = 0; k0 < INDIM; k0 += 4) {
      v2f a, b;
      a[0] = xs[ln][k0 + 2 * half];
      a[1] = xs[ln][k0 + 2 * half + 1];
      const float* wp = W + (size_t)(k0 + 2 * half) * HD + col0 + ln;
      b[0] = wp[0];
      b[1] = wp[HD];
      c = wmma_f32_16x16x4(a, b, c);
    }
    float* o = outs[m];
#pragma unroll
    for (int r = 0; r < 8; ++r)  // C layout: VGPR r -> M = r + 8*half
      o[(size_t)(m0 + r + 8 * half) * HD + col0 + ln] = c[r];
  }
}

// -------- K2: fused edge GEMM + conn + softmax-weight + scatter adds -------
// 64 edges per block, one wave per head.  Each weight B-fragment feeds 4
// WMMAs (one per 16-edge M-tile) -> ~0.5 vmem per wmma in the GEMM loop.
__global__ __launch_bounds__(256) void edge_fused_kernel(
    const float* __restrict__ edge_attr, const int* __restrict__ ei,  // [2,E]
    const float* __restrict__ Weq, const float* __restrict__ beq,
    const float* __restrict__ Nq, const float* __restrict__ Nk,
    const float* __restrict__ Nv, const float* __restrict__ Aw,
    float* __restrict__ e_out, float* __restrict__ ssum,
    float* __restrict__ agg1, float* __restrict__ agg2, int E) {
  __shared__ float xs[EPB][LDSPITCH];
  const int tid = threadIdx.x;
  const int e0 = blockIdx.x * EPB;
  {
    const int row = tid >> 2;         // 4 threads per row
    const int c0 = (tid & 3) * 32;    // 32 consecutive floats each
    const float* src = edge_attr + (size_t)(e0 + row) * INDIM + c0;
    float* dstp = &xs[row][c0];
#pragma unroll
    for (int j = 0; j < 32; j += 4) {
      const float4 v = *(const float4*)(src + j);
      dstp[j + 0] = v.x;
      dstp[j + 1] = v.y;
      dstp[j + 2] = v.z;
      dstp[j + 3] = v.w;
    }
  }
  __syncthreads();

  const int lane = tid & 31;
  const int half = lane >> 4;
  const int ln = lane & 15;
  const int h = tid >> 5;  // one wave per head
  const int col0 = h * HDIM;

  const float bias = beq[col0 + ln];
  v8f acc[4];
#pragma unroll
  for (int t = 0; t < 4; ++t)
#pragma unroll
    for (int r = 0; r < 8; ++r) acc[t][r] = bias;

#pragma unroll
  for (int k0 = 0; k0 < INDIM; k0 += 4) {
    v2f b;
    const float* wp = Weq + (size_t)(k0 + 2 * half) * HD + col0 + ln;
    b[0] = wp[0];
    b[1] = wp[HD];
#pragma unroll
    for (int t = 0; t < 4; ++t) {
      v2f a;
      a[0] = xs[t * 16 + ln][k0 + 2 * half];
      a[1] = xs[t * 16 + ln][k0 + 2 * half + 1];
      acc[t] = wmma_f32_16x16x4(a, b, acc[t]);
    }
  }

  const float aw = Aw[ln * N_HEADS + h];  // Aw[d, h, 0]
#pragma unroll
  for (int t = 0; t < 4; ++t) {
#pragma unroll
    for (int r = 0; r < 8; ++r) {
      const int er = e0 + t * 16 + r + 8 * half;
      const int s = ei[er];
      const int d = ei[E + er];
      float cn = acc[t][r] + Nk[(size_t)s * HD + col0 + ln] +
                 Nq[(size_t)d * HD + col0 + ln];
      cn = fmaxf(cn, 0.0f);                      // relu -> conn
      e_out[(size_t)er * HD + col0 + ln] = cn;   // e_out = conn
      float sc = cn * aw;                        // reduce over d (16 lanes)
      sc += __shfl_xor(sc, 8, 32);
      sc += __shfl_xor(sc, 4, 32);
      sc += __shfl_xor(sc, 2, 32);
      sc += __shfl_xor(sc, 1, 32);
      sc = fminf(fmaxf(sc, -CLAMP_V), CLAMP_V);
      const float p = expf(sc);                  // safe: sc in [-5, 5]
      const float vv = Nv[(size_t)s * HD + col0 + ln];
      atomicAdd(&agg1[(size_t)d * HD + col0 + ln], p * vv);
      atomicAdd(&agg2[(size_t)d * HD + col0 + ln], p * cn);
      if (ln == 0) atomicAdd(&ssum[(size_t)d * N_HEADS + h], p);
    }
  }
}

// --------- K3: per-head 16x16 Ew contraction (WMMA) + normalization --------
__global__ __launch_bounds__(256) void node_out_kernel(
    const float* __restrict__ agg1, const float* __restrict__ agg2,
    const float* __restrict__ ssum, const float* __restrict__ Ew,
    float* __restrict__ n_out) {
  const int tid = threadIdx.x;
  const int n0 = blockIdx.x * 16;
  const int lane = tid & 31;
  const int half = lane >> 4;
  const int ln = lane & 15;
  const int h = tid >> 5;
  const int col0 = h * HDIM;

  v8f c;  // accumulator pre-loaded with agg1 -> one MAC chain for agg1+agg3
#pragma unroll
  for (int r = 0; r < 8; ++r)
    c[r] = agg1[(size_t)(n0 + r + 8 * half) * HD + col0 + ln];
#pragma unroll
  for (int k0 = 0; k0 < HDIM; k0 += 4) {  // agg3 = agg2 @ Ew[:,h,:]
    v2f a, b;
    const float* ap = agg2 + (size_t)(n0 + ln) * HD + col0 + k0 + 2 * half;
    a[0] = ap[0];
    a[1] = ap[1];
    const float* bp = Ew + (size_t)(k0 + 2 * half) * HD + col0 + ln;
    b[0] = bp[0];
    b[1] = bp[HD];
    c = wmma_f32_16x16x4(a, b, c);
  }
#pragma unroll
  for (int r = 0; r < 8; ++r) {
    const int n = n0 + r + 8 * half;
    const float den = ssum[(size_t)n * N_HEADS + h] + 1e-16f;
    n_out[(size_t)n * HD + col0 + ln] = c[r] / den;
  }
}

// ------------------------------ init fill ----------------------------------
__global__ void fill_f32_kernel(float* p, float v, long n) {
  const long i = (long)blockIdx.x * 256 + threadIdx.x;
  if (i < n) p[i] = v;
}

extern "C" void kernel_launch(void* const* d_in, const int* in_sizes, int n_in,
                              void* d_out, int out_size, void* d_ws,
                              size_t ws_size, hipStream_t stream) {
  const float* x = (const float*)d_in[0];
  const float* edge_attr = (const float*)d_in[1];
  const int* ei = (const int*)d_in[2];  // jax default x64=off -> int32
  const float* Wq = (const float*)d_in[4];
  const float* bq = (const float*)d_in[5];
  const float* Wk = (const float*)d_in[6];
  const float* bk = (const float*)d_in[7];
  const float* Wv = (const float*)d_in[8];
  const float* bv = (const float*)d_in[9];
  const float* Weq = (const float*)d_in[10];
  const float* beq = (const float*)d_in[11];
  const float* Aw = (const float*)d_in[12];
  const float* Ew = (const float*)d_in[13];

  const int N = in_sizes[0] / INDIM;  // 50000 (multiple of 16)
  const int E = in_sizes[1] / INDIM;  // 800000 (multiple of EPB=64)

  float* n_out = (float*)d_out;
  float* e_out = n_out + (size_t)N * HD;

  // Workspace layout (f32 units); zero-region [ssum|agg1|agg2] is contiguous.
  float* ws = (float*)d_ws;
  size_t off = 0;
  float* Nq = ws + off;   off += (size_t)N * HD;
  float* Nk = ws + off;   off += (size_t)N * HD;
  float* Nv = ws + off;   off += (size_t)N * HD;
  float* ssum = ws + off; off += (size_t)N * N_HEADS;
  float* agg1 = ws + off; off += (size_t)N * HD;
  float* agg2 = ws + off; off += (size_t)N * HD;

  const long n_zero = (long)N * (N_HEADS + HD + HD);
  fill_f32_kernel<<<(unsigned)((n_zero + 255) / 256), 256, 0, stream>>>(
      ssum, 0.0f, n_zero);

  qkv_gemm_kernel<<<N / 16, 256, 0, stream>>>(x, Wq, bq, Wk, bk, Wv, bv, Nq,
                                              Nk, Nv);
  edge_fused_kernel<<<E / EPB, 256, 0, stream>>>(edge_attr, ei, Weq, beq, Nq,
                                                 Nk, Nv, Aw, e_out, ssum, agg1,
                                                 agg2, E);
  node_out_kernel<<<N / 16, 256, 0, stream>>>(agg1, agg2, ssum, Ew, n_out);
}